// EnhancedPathAwareECA_14164802142826
// MI455X (gfx1250) — compile-verified
//
#include <hip/hip_runtime.h>
#include <math.h>

typedef __attribute__((ext_vector_type(2))) float v2f;
typedef __attribute__((ext_vector_type(4))) float v4f;
typedef __attribute__((ext_vector_type(8))) float v8f;

#define B_   16
#define P_   8
#define D_   128
#define L_   4096
#define ROWS (B_ * P_ * D_)   // 16384 rows of length L_

// ---------------------------------------------------------------------------
// Kernel A: y[b,p,d] = mean over l of x[b, p*D+d, l].  One block per row.
// Streams 256 MiB at full bandwidth with b128 loads.
// ---------------------------------------------------------------------------
__global__ __launch_bounds__(256) void row_mean_kernel(const float* __restrict__ x,
                                                       float* __restrict__ y) {
    __shared__ float red[256];
    const int row = blockIdx.x;
    const v4f* xr = (const v4f*)(x + (size_t)row * L_);
    float s = 0.f;
#pragma unroll
    for (int j = 0; j < 4; ++j) {
        v4f v = xr[threadIdx.x + j * 256];
        s += v.x + v.y + v.z + v.w;
    }
    red[threadIdx.x] = s;
    __syncthreads();
    for (int off = 128; off > 0; off >>= 1) {
        if (threadIdx.x < off) red[threadIdx.x] += red[threadIdx.x + off];
        __syncthreads();
    }
    if (threadIdx.x == 0) y[row] = red[0] * (1.0f / (float)L_);
}

// ---------------------------------------------------------------------------
// Kernel B: ECA convs (k=5, k=9, same zero padding, cross-correlation) over
// the D dimension, sigmoid combine -> attn; block-mean of attn -> cross.
// One block per (b,p); 128 threads, one per d.
// ---------------------------------------------------------------------------
__global__ __launch_bounds__(128) void attn_kernel(const float* __restrict__ y,
                                                   const float* __restrict__ w1,
                                                   const float* __restrict__ w2,
                                                   const float* __restrict__ cw,
                                                   const float* __restrict__ cb,
                                                   float* __restrict__ attn,
                                                   float* __restrict__ cross) {
    __shared__ float yl[D_];
    __shared__ float red[D_];
    const int bp = blockIdx.x;      // b*8 + p
    const int p  = bp & 7;
    const int d  = threadIdx.x;
    yl[d] = y[bp * D_ + d];
    __syncthreads();

    float s1 = 0.f;
#pragma unroll
    for (int j = 0; j < 5; ++j) {
        int t = d + j - 2;
        float v = (t >= 0 && t < D_) ? yl[t] : 0.f;
        s1 += w1[p * 5 + j] * v;
    }
    float s2 = 0.f;
#pragma unroll
    for (int j = 0; j < 9; ++j) {
        int t = d + j - 4;
        float v = (t >= 0 && t < D_) ? yl[t] : 0.f;
        s2 += w2[p * 9 + j] * v;
    }
    float z = cw[0] * s1 + cw[1] * s2 + cb[0];
    float a = 1.0f / (1.0f + expf(-z));
    attn[bp * D_ + d] = a;

    red[d] = a;
    __syncthreads();
    for (int off = 64; off > 0; off >>= 1) {
        if (d < off) red[d] += red[d + off];
        __syncthreads();
    }
    if (d == 0) cross[bp] = red[0] * (1.0f / (float)D_);
}

// ---------------------------------------------------------------------------
// Kernel C: LayerNorm over P, MLP 8->16 (exact GELU) -> 16->8, sigmoid gate.
// Single wave32; matmuls via V_WMMA_F32_16X16X4_F32.
//
// A-matrix 16x4 f32 layout (ISA 7.12.2): lanes 0-15 / 16-31 hold M=0..15;
//   VGPR0 = K=(0|2), VGPR1 = K=(1|3) depending on lane half.
// B-matrix 4x16: rows (K) striped like C/D rows: VGPR0 = K=(0|2), VGPR1 =
//   K=(1|3); N = lane%16.
// C/D 16x16: VGPRv = row (v | v+8) by lane half, N = lane%16.
// ---------------------------------------------------------------------------
__global__ __launch_bounds__(32) void mlp_kernel(const float* __restrict__ cross,
                                                 const float* __restrict__ ln_g,
                                                 const float* __restrict__ ln_b,
                                                 const float* __restrict__ W1,
                                                 const float* __restrict__ b1,
                                                 const float* __restrict__ W2,
                                                 const float* __restrict__ b2,
                                                 float* __restrict__ gate) {
    __shared__ float sh_h[16][8];     // normalized cross
    __shared__ float sh_h1[16][16];   // hidden after GELU
    const int lane = threadIdx.x;

    // LayerNorm over P=8 per batch row (biased variance, eps=1e-5)
    if (lane < 16) {
        float v[8];
        float mu = 0.f;
#pragma unroll
        for (int i = 0; i < 8; ++i) { v[i] = cross[lane * 8 + i]; mu += v[i]; }
        mu *= 0.125f;
        float var = 0.f;
#pragma unroll
        for (int i = 0; i < 8; ++i) { float t = v[i] - mu; var += t * t; }
        var *= 0.125f;
        float inv = rsqrtf(var + 1e-5f);
#pragma unroll
        for (int i = 0; i < 8; ++i)
            sh_h[lane][i] = (v[i] - mu) * inv * ln_g[i] + ln_b[i];
    }
    __syncthreads();

    const int M  = lane & 15;               // also the N column for B/C/D
    const int kh = (lane < 16) ? 0 : 2;     // K-half offset for this lane group

    // Layer 1: h(16x8) @ W1(8x16)  -> two K=4 WMMA steps
    v8f acc = {};
#pragma unroll
    for (int kb = 0; kb < 8; kb += 4) {
        v2f amat, bmat;
        amat.x = sh_h[M][kb + kh];
        amat.y = sh_h[M][kb + kh + 1];
        bmat.x = W1[(kb + kh) * 16 + M];        // W1 is (8,16) row-major
        bmat.y = W1[(kb + kh + 1) * 16 + M];
        acc = __builtin_amdgcn_wmma_f32_16x16x4_f32(
            false, amat, false, bmat, (short)0, acc, false, false);
    }

    // bias + exact GELU, scatter D layout into (16x16) LDS tile
#pragma unroll
    for (int v = 0; v < 8; ++v) {
        int m = v + ((lane < 16) ? 0 : 8);
        float t = acc[v] + b1[M];
        sh_h1[m][M] = 0.5f * t * (1.0f + erff(t * 0.70710678118654752f));
    }
    __syncthreads();

    // Layer 2: h1(16x16) @ W2(16x8, zero-padded to 16 cols) -> four K=4 steps
    v8f acc2 = {};
#pragma unroll
    for (int kb = 0; kb < 16; kb += 4) {
        v2f amat, bmat;
        amat.x = sh_h1[M][kb + kh];
        amat.y = sh_h1[M][kb + kh + 1];
        bmat.x = (M < 8) ? W2[(kb + kh) * 8 + M] : 0.f;     // W2 is (16,8)
        bmat.y = (M < 8) ? W2[(kb + kh + 1) * 8 + M] : 0.f;
        acc2 = __builtin_amdgcn_wmma_f32_16x16x4_f32(
            false, amat, false, bmat, (short)0, acc2, false, false);
    }

    // sigmoid gate, keep only the valid 8 columns
#pragma unroll
    for (int v = 0; v < 8; ++v) {
        int m = v + ((lane < 16) ? 0 : 8);
        if (M < 8) {
            float t = acc2[v] + b2[M];
            gate[m * 8 + M] = 1.0f / (1.0f + expf(-t));
        }
    }
}

// ---------------------------------------------------------------------------
// Kernel D: out[b,p,d,:] = x[b,p,d,:] * attn[b,p,d] * gate[b,p].
// One block per row; uniform per-block scalar; nontemporal b128 stream
// (write-once output, don't pollute L2/WGP$ on the way out).
// ---------------------------------------------------------------------------
__global__ __launch_bounds__(256) void scale_kernel(const float* __restrict__ x,
                                                    const float* __restrict__ attn,
                                                    const float* __restrict__ gate,
                                                    float* __restrict__ out) {
    const int row = blockIdx.x;
    const float s = attn[row] * gate[row >> 7];
    const v4f* xr = (const v4f*)(x + (size_t)row * L_);
    v4f* orow = (v4f*)(out + (size_t)row * L_);
#pragma unroll
    for (int j = 0; j < 4; ++j) {
        int idx = threadIdx.x + j * 256;
        v4f v = __builtin_nontemporal_load(&xr[idx]);
        v4f r = v * s;
        __builtin_nontemporal_store(r, &orow[idx]);
    }
}

// ---------------------------------------------------------------------------
extern "C" void kernel_launch(void* const* d_in, const int* in_sizes, int n_in,
                              void* d_out, int out_size, void* d_ws, size_t ws_size,
                              hipStream_t stream) {
    const float* x         = (const float*)d_in[0];
    const float* conv1_w   = (const float*)d_in[1];   // (8,5)
    const float* conv2_w   = (const float*)d_in[2];   // (8,9)
    const float* combine_w = (const float*)d_in[3];   // (2,)
    const float* combine_b = (const float*)d_in[4];   // scalar
    const float* ln_g      = (const float*)d_in[5];   // (8,)
    const float* ln_b      = (const float*)d_in[6];   // (8,)
    const float* W1        = (const float*)d_in[7];   // (8,16)
    const float* b1        = (const float*)d_in[8];   // (16,)
    const float* W2        = (const float*)d_in[9];   // (16,8)
    const float* b2        = (const float*)d_in[10];  // (8,)
    float* out = (float*)d_out;

    float* y     = (float*)d_ws;        // ROWS floats
    float* attn  = y + ROWS;            // ROWS floats
    float* cross = attn + ROWS;         // 128 floats
    float* gate  = cross + B_ * P_;     // 128 floats

    row_mean_kernel<<<ROWS, 256, 0, stream>>>(x, y);
    attn_kernel<<<B_ * P_, 128, 0, stream>>>(y, conv1_w, conv2_w,
                                             combine_w, combine_b, attn, cross);
    mlp_kernel<<<1, 32, 0, stream>>>(cross, ln_g, ln_b, W1, b1, W2, b2, gate);
    scale_kernel<<<ROWS, 256, 0, stream>>>(x, attn, gate, out);
}